// AttentionLSTMDecoder_18614388261437
// MI455X (gfx1250) — compile-verified
//
#include <hip/hip_runtime.h>
#include <math.h>

#define B_ 64
#define S_ 512
#define T_ 64
#define E_ 512
#define V_ 128
#define H_ 512
#define G_ (4 * H_)
#define CI_ (E_ + V_)
#define SCALE_ 0.04419417382415922f  /* 1/sqrt(512) */

typedef float v2f __attribute__((ext_vector_type(2)));
typedef float v8f __attribute__((ext_vector_type(8)));

// ---------------------------------------------------------------------------
// WMMA helpers (fp32 16x16x4, wave32).
// A-fragment layout (16x4, 32-bit): lane L holds row M=L%16; VGPR v holds
// K = v + 2*(L/16). B-fragment (4x16) mirrors it: lane L holds col N=L%16,
// VGPR v holds K = v + 2*(L/16). For a row-major matrix, both tile loads are
// "row = base + L%16, cols = e + 2*(L/16) + {0,1}" -> one 8-byte load.
// ---------------------------------------------------------------------------
__device__ __forceinline__ v2f ld_frag(const float* __restrict__ src, int ld) {
  int lane = threadIdx.x & 31;
  return *(const v2f*)(src + (size_t)(lane & 15) * ld + ((lane >> 4) << 1));
}

__device__ __forceinline__ v8f wmma_f32(v2f a, v2f b, v8f c) {
  return __builtin_amdgcn_wmma_f32_16x16x4_f32(false, a, false, b, (short)0, c,
                                               false, false);
}

// C/D 16x16 f32: VGPR r, lane L -> M = r + 8*(L/16), N = L%16.
__device__ __forceinline__ void st_tile(float* __restrict__ dst, int ld, v8f c) {
  int lane = threadIdx.x & 31;
  int n = lane & 15;
  int m0 = (lane >> 4) << 3;
#pragma unroll
  for (int r = 0; r < 8; ++r) dst[(size_t)(m0 + r) * ld + n] = c[r];
}

__device__ __forceinline__ v8f bias_acc(const float* __restrict__ bias, int n0) {
  int lane = threadIdx.x & 31;
  float bv = bias[n0 + (lane & 15)];
  v8f acc;
#pragma unroll
  for (int r = 0; r < 8; ++r) acc[r] = bv;
  return acc;
}

// ---------------------------------------------------------------------------
// Fused K/V projection: K[b,s,f] = mem[b,s,:]·Wk[f,:] + bk[f] (same for V).
// One A-fragment feeds two WMMAs (K and V weight tiles).
// grid = (BS/16, E/64), block = 128 (4 waves, one N-tile each).
// ---------------------------------------------------------------------------
__global__ void kv_proj_kernel(const float* __restrict__ mem,
                               const float* __restrict__ Wk,
                               const float* __restrict__ bk,
                               const float* __restrict__ Wv,
                               const float* __restrict__ bv,
                               float* __restrict__ Ko, float* __restrict__ Vo) {
  int wave = threadIdx.x >> 5;
  int mt = blockIdx.x;
  int nt = blockIdx.y * 4 + wave;
  v8f accK = bias_acc(bk, nt << 4);
  v8f accV = bias_acc(bv, nt << 4);
  const float* arow = mem + (size_t)mt * 16 * E_;
  const float* wkr = Wk + (size_t)nt * 16 * E_;
  const float* wvr = Wv + (size_t)nt * 16 * E_;
  for (int e = 0; e < E_; e += 8) {
#pragma unroll
    for (int u = 0; u < 2; ++u) {
      v2f a = ld_frag(arow + e + u * 4, E_);
      v2f fk = ld_frag(wkr + e + u * 4, E_);
      v2f fv = ld_frag(wvr + e + u * 4, E_);
      accK = wmma_f32(a, fk, accK);
      accV = wmma_f32(a, fv, accV);
    }
  }
  st_tile(Ko + (size_t)mt * 16 * E_ + (nt << 4), E_, accK);
  st_tile(Vo + (size_t)mt * 16 * E_ + (nt << 4), E_, accV);
}

// ---------------------------------------------------------------------------
// Generic C = A(MxK) · W(NxK)^T + bias, all row-major, ldc-strided output.
// grid = (M/16, N/64), block = 128.
// ---------------------------------------------------------------------------
__global__ void gemm_bias_kernel(const float* __restrict__ A, int lda,
                                 const float* __restrict__ W, int ldw,
                                 const float* __restrict__ bias,
                                 float* __restrict__ C, int ldc, int K) {
  int wave = threadIdx.x >> 5;
  int mt = blockIdx.x;
  int nt = blockIdx.y * 4 + wave;
  v8f acc = bias_acc(bias, nt << 4);
  const float* arow = A + (size_t)mt * 16 * lda;
  const float* wrow = W + (size_t)nt * 16 * ldw;
  for (int e = 0; e < K; e += 16) {
#pragma unroll
    for (int u = 0; u < 4; ++u) {
      v2f a = ld_frag(arow + e + u * 4, lda);
      v2f b = ld_frag(wrow + e + u * 4, ldw);
      acc = wmma_f32(a, b, acc);
    }
  }
  st_tile(C + (size_t)mt * 16 * ldc + (nt << 4), ldc, acc);
}

// ---------------------------------------------------------------------------
// gates = [ctxo | x_t]·W_ih^T + h·W_hh^T + b_ih + b_hh, three K segments fused.
// grid = (B/16, G/64), block = 128.
// ---------------------------------------------------------------------------
__global__ void gates_kernel(const float* __restrict__ ctxo,
                             const float* __restrict__ xt, int xt_ld,
                             const float* __restrict__ h,
                             const float* __restrict__ W_ih,
                             const float* __restrict__ b_ih,
                             const float* __restrict__ W_hh,
                             const float* __restrict__ b_hh,
                             float* __restrict__ gates) {
  int wave = threadIdx.x >> 5;
  int mt = blockIdx.x;
  int nt = blockIdx.y * 4 + wave;
  int lane = threadIdx.x & 31;
  int n = (nt << 4) + (lane & 15);
  float bsum = b_ih[n] + b_hh[n];
  v8f acc;
#pragma unroll
  for (int r = 0; r < 8; ++r) acc[r] = bsum;

  const float* w1 = W_ih + (size_t)nt * 16 * CI_;
  const float* crow = ctxo + (size_t)mt * 16 * E_;
  for (int e = 0; e < E_; e += 16) {
#pragma unroll
    for (int u = 0; u < 4; ++u)
      acc = wmma_f32(ld_frag(crow + e + u * 4, E_),
                     ld_frag(w1 + e + u * 4, CI_), acc);
  }
  const float* xrow = xt + (size_t)mt * 16 * xt_ld;
  for (int e = 0; e < V_; e += 16) {
#pragma unroll
    for (int u = 0; u < 4; ++u)
      acc = wmma_f32(ld_frag(xrow + e + u * 4, xt_ld),
                     ld_frag(w1 + E_ + e + u * 4, CI_), acc);
  }
  const float* w2 = W_hh + (size_t)nt * 16 * H_;
  const float* hrow = h + (size_t)mt * 16 * H_;
  for (int e = 0; e < H_; e += 16) {
#pragma unroll
    for (int u = 0; u < 4; ++u)
      acc = wmma_f32(ld_frag(hrow + e + u * 4, H_),
                     ld_frag(w2 + e + u * 4, H_), acc);
  }
  st_tile(gates + (size_t)mt * 16 * G_ + (nt << 4), G_, acc);
}

// ---------------------------------------------------------------------------
// scores[b,s] = scale * q[b,:]·K[b,s,:]. One wave per (b,s); float4 loads,
// wave-shuffle reduction. K resides in L2 (128 MB K+V < 192 MB L2).
// ---------------------------------------------------------------------------
__global__ void attn_scores_kernel(const float* __restrict__ q,
                                   const float* __restrict__ Kmat,
                                   float* __restrict__ scores) {
  int w = blockIdx.x * 8 + (threadIdx.x >> 5);
  int lane = threadIdx.x & 31;
  int b = w >> 9;
  int s = w & 511;
  const float* kp = Kmat + ((size_t)b * S_ + s) * E_;
  const float* qp = q + (size_t)b * E_;
  float sum = 0.f;
#pragma unroll
  for (int i = 0; i < 4; ++i) {
    int e = i * 128 + lane * 4;
    float4 kv = *(const float4*)(kp + e);
    float4 qv = *(const float4*)(qp + e);
    sum += kv.x * qv.x + kv.y * qv.y + kv.z * qv.z + kv.w * qv.w;
  }
#pragma unroll
  for (int off = 16; off; off >>= 1) sum += __shfl_xor(sum, off, 32);
  if (lane == 0) scores[w] = sum * SCALE_;
}

// softmax over S=512, one workgroup (256 threads = 8 waves) per batch row.
__global__ void softmax_kernel(float* __restrict__ scores) {
  __shared__ float red[8];
  float* p = scores + (size_t)blockIdx.x * S_;
  int t = threadIdx.x;
  float x0 = p[t], x1 = p[t + 256];
  float m = fmaxf(x0, x1);
#pragma unroll
  for (int off = 16; off; off >>= 1) m = fmaxf(m, __shfl_xor(m, off, 32));
  if ((t & 31) == 0) red[t >> 5] = m;
  __syncthreads();
  float gm = red[0];
#pragma unroll
  for (int i = 1; i < 8; ++i) gm = fmaxf(gm, red[i]);
  float e0 = expf(x0 - gm), e1 = expf(x1 - gm);
  float ssum = e0 + e1;
#pragma unroll
  for (int off = 16; off; off >>= 1) ssum += __shfl_xor(ssum, off, 32);
  __syncthreads();
  if ((t & 31) == 0) red[t >> 5] = ssum;
  __syncthreads();
  float gs = 0.f;
#pragma unroll
  for (int i = 0; i < 8; ++i) gs += red[i];
  float inv = 1.0f / gs;
  p[t] = e0 * inv;
  p[t + 256] = e1 * inv;
}

// ctx[b,e] = sum_s attn[b,s]*V[b,s,e]. Wave per (b, 32-wide e-chunk); lane
// owns one e -> coalesced V rows, scalar attn broadcast.
__global__ void ctx_kernel(const float* __restrict__ attn,
                           const float* __restrict__ Vmat,
                           float* __restrict__ ctx) {
  int w = blockIdx.x * 8 + (threadIdx.x >> 5);
  int lane = threadIdx.x & 31;
  int b = w >> 4;
  int e = ((w & 15) << 5) + lane;
  const float* ap = attn + (size_t)b * S_;
  const float* vp = Vmat + (size_t)b * S_ * E_ + e;
  float sum = 0.f;
  for (int s = 0; s < S_; s += 4) {
    sum += ap[s] * vp[(size_t)s * E_];
    sum += ap[s + 1] * vp[(size_t)(s + 1) * E_];
    sum += ap[s + 2] * vp[(size_t)(s + 2) * E_];
    sum += ap[s + 3] * vp[(size_t)(s + 3) * E_];
  }
  ctx[(size_t)b * E_ + e] = sum;
}

__global__ void lstm_kernel(const float* __restrict__ gates,
                            float* __restrict__ c, float* __restrict__ h) {
  int idx = blockIdx.x * blockDim.x + threadIdx.x;  // B*H
  int b = idx >> 9, n = idx & 511;
  const float* g = gates + (size_t)b * G_;
  float gi = g[n], gf = g[H_ + n], gg = g[2 * H_ + n], go = g[3 * H_ + n];
  float si = 1.f / (1.f + expf(-gi));
  float sf = 1.f / (1.f + expf(-gf));
  float so = 1.f / (1.f + expf(-go));
  float cn = sf * c[idx] + si * tanhf(gg);
  c[idx] = cn;
  h[idx] = so * tanhf(cn);
}

__global__ void zero_kernel(float* __restrict__ p, int n) {
  int i = blockIdx.x * blockDim.x + threadIdx.x;
  if (i < n) p[i] = 0.f;
}

// ---------------------------------------------------------------------------
extern "C" void kernel_launch(void* const* d_in, const int* in_sizes, int n_in,
                              void* d_out, int out_size, void* d_ws,
                              size_t ws_size, hipStream_t stream) {
  const float* mem  = (const float*)d_in[0];
  const float* tgt  = (const float*)d_in[1];
  const float* Wq   = (const float*)d_in[2];
  const float* bq   = (const float*)d_in[3];
  const float* Wk   = (const float*)d_in[4];
  const float* bk   = (const float*)d_in[5];
  const float* Wv   = (const float*)d_in[6];
  const float* bv   = (const float*)d_in[7];
  const float* Wo   = (const float*)d_in[8];
  const float* bo   = (const float*)d_in[9];
  const float* W_ih = (const float*)d_in[10];
  const float* b_ih = (const float*)d_in[11];
  const float* W_hh = (const float*)d_in[12];
  const float* b_hh = (const float*)d_in[13];
  const float* Wcls = (const float*)d_in[14];
  const float* bcls = (const float*)d_in[15];
  float* out = (float*)d_out;

  float* ws = (float*)d_ws;
  float* Kw = ws;                              // B*S*E
  float* Vw = Kw + (size_t)B_ * S_ * E_;       // B*S*E
  float* h  = Vw + (size_t)B_ * S_ * E_;       // B*H
  float* c  = h + B_ * H_;                     // B*H
  float* q  = c + B_ * H_;                     // B*E
  float* sc = q + B_ * E_;                     // B*S
  float* ctx = sc + B_ * S_;                   // B*E
  float* ctxo = ctx + B_ * E_;                 // B*E
  float* gates = ctxo + B_ * E_;               // B*4H

  zero_kernel<<<(2 * B_ * H_) / 256, 256, 0, stream>>>(h, 2 * B_ * H_);
  kv_proj_kernel<<<dim3(B_ * S_ / 16, E_ / 64), 128, 0, stream>>>(
      mem, Wk, bk, Wv, bv, Kw, Vw);

  for (int t = 0; t < T_; ++t) {
    // q = h·Wq^T + bq
    gemm_bias_kernel<<<dim3(B_ / 16, E_ / 64), 128, 0, stream>>>(
        h, H_, Wq, E_, bq, q, E_, E_);
    attn_scores_kernel<<<(B_ * S_) / 8, 256, 0, stream>>>(q, Kw, sc);
    softmax_kernel<<<B_, 256, 0, stream>>>(sc);
    ctx_kernel<<<(B_ * (E_ / 32)) / 8, 256, 0, stream>>>(sc, Vw, ctx);
    // ctxo = ctx·Wo^T + bo
    gemm_bias_kernel<<<dim3(B_ / 16, E_ / 64), 128, 0, stream>>>(
        ctx, E_, Wo, E_, bo, ctxo, E_, E_);
    gates_kernel<<<dim3(B_ / 16, G_ / 64), 128, 0, stream>>>(
        ctxo, tgt + (size_t)t * V_, T_ * V_, h, W_ih, b_ih, W_hh, b_hh, gates);
    lstm_kernel<<<(B_ * H_) / 256, 256, 0, stream>>>(gates, c, h);
    // out[:, t, :] = h_new·Wcls^T + bcls (strided store into (B,T,V))
    gemm_bias_kernel<<<dim3(B_ / 16, V_ / 64), 128, 0, stream>>>(
        h, H_, Wcls, H_, bcls, out + (size_t)t * V_, T_ * V_, H_);
  }
}